// QueryTokenProbe_807453852079
// MI455X (gfx1250) — compile-verified
//
#include <hip/hip_runtime.h>

// Problem constants (match reference setup_inputs()).
constexpr int Bn = 32;
constexpr int Sn = 2048;
constexpr int Dn = 2048;
constexpr int Kn = 6;
constexpr int NROWS = Bn * Kn;        // 192 gathered rows
constexpr int TILES = NROWS / 16;     // 12 row tiles of 16

typedef float v2f __attribute__((ext_vector_type(2)));
typedef float v8f __attribute__((ext_vector_type(8)));

__device__ __forceinline__ int wave_max_i32(int v) {
#pragma unroll
    for (int off = 16; off > 0; off >>= 1)
        v = max(v, __shfl_xor(v, off, 32));
    return v;
}

// ---------------------------------------------------------------------------
// Kernel 1: per-sample scan.  One block per batch sample, 256 threads.
// Computes final gather position per (b,k) and has_valid flag per b.
// ---------------------------------------------------------------------------
__global__ void k_positions(const int* __restrict__ ids,
                            const int* __restrict__ amask,
                            const int* __restrict__ tok,
                            int* __restrict__ pos_out,     // [B*K]
                            int* __restrict__ hasv_out) {  // [B]
    const int b   = blockIdx.x;
    const int tid = threadIdx.x;

    __shared__ int s_max[1 + Kn];   // [0]=last_valid, [1..6]=last_match per token
    if (tid < 1 + Kn) s_max[tid] = -1;
    __syncthreads();

    int tk[Kn];
#pragma unroll
    for (int k = 0; k < Kn; ++k) tk[k] = tok[k];

    const int* idrow = ids   + (long)b * Sn;
    const int* mkrow = amask + (long)b * Sn;

    int lv = -1;
    int lm[Kn] = {-1, -1, -1, -1, -1, -1};
    // s strictly increases per thread, so the final assignment is the max.
    for (int s = tid; s < Sn; s += 256) {
        if (mkrow[s] > 0) {
            lv = s;
            const int id = idrow[s];
#pragma unroll
            for (int k = 0; k < Kn; ++k)
                if (id == tk[k]) lm[k] = s;
        }
    }

    lv = wave_max_i32(lv);
#pragma unroll
    for (int k = 0; k < Kn; ++k) lm[k] = wave_max_i32(lm[k]);

    if ((tid & 31) == 0) {
        atomicMax(&s_max[0], lv);
#pragma unroll
        for (int k = 0; k < Kn; ++k) atomicMax(&s_max[1 + k], lm[k]);
    }
    __syncthreads();

    if (tid == 0) {
        const int last_valid = s_max[0];
        hasv_out[b] = (last_valid >= 0) ? 1 : 0;
#pragma unroll
        for (int k = 0; k < Kn; ++k) {
            int p = (s_max[1 + k] >= 0) ? s_max[1 + k] : last_valid;
            pos_out[b * Kn + k] = max(p, 0);   // clip(final_pos, 0, S-1)
        }
    }
}

// ---------------------------------------------------------------------------
// Kernel 2: gathered GEMM via V_WMMA_F32_16X16X4_F32 (exact f32 FMA path).
//   A = H  (192 x 2048 gathered rows), B = W^T zero-padded to 16 cols.
// One block per 16-row tile; 8 waves each accumulate a K=256 slice
// (64 chained wmma ops), then a fixed-order LDS reduction.
// A-fragment (ISA 16x4 f32 layout): lane%16 = M row; K pair offset 2*(lane/16).
// B-fragment mirrors with N across lanes.
// C/D layout: c[vr] = C[M = vr + 8*(lane/16)][N = lane%16].
// ---------------------------------------------------------------------------
__global__ void k_logits(const float* __restrict__ hs,
                         const float* __restrict__ W,
                         const float* __restrict__ bias,
                         const int*   __restrict__ pos,
                         const int*   __restrict__ hasv,
                         float* __restrict__ out) {
    const int tile = blockIdx.x;              // 0..11
    const int lane = threadIdx.x & 31;
    const int wave = threadIdx.x >> 5;        // 0..7
    const int m    = lane & 15;
    const int half = lane >> 4;

    // A: this lane's gathered hidden-state row.
    const int  r   = tile * 16 + m;           // 0..191  (row = b*K + k)
    const int  bb  = r / Kn;
    const long arow = ((long)bb * Sn + pos[r]) * Dn;
    const float* aptr = hs + arow + 2 * half;

    // B: column n = lane%16 -> W row n (zero for n >= 6, via multiplier to
    // keep EXEC all-ones as WMMA requires).
    const float sel  = (m < Kn) ? 1.0f : 0.0f;
    const int   wrow = (m < Kn) ? m : 0;
    const float* bptr = W + (long)wrow * Dn + 2 * half;

    const int kbase = wave * (Dn / 8);        // 256-wide K slice per wave

    v8f c = {};
#pragma unroll 4
    for (int k0 = 0; k0 < Dn / 8; k0 += 4) {
        v2f a  = *(const v2f*)(aptr + kbase + k0);
        v2f bv = *(const v2f*)(bptr + kbase + k0);
        bv *= sel;
        c = __builtin_amdgcn_wmma_f32_16x16x4_f32(
                /*neg_a=*/false, a, /*neg_b=*/false, bv,
                /*c_mod=*/(short)0, c, /*reuse_a=*/false, /*reuse_b=*/false);
    }

    // Deterministic fixed-order cross-wave reduction.
    __shared__ float red[8][32][8];           // 8 KB
#pragma unroll
    for (int i = 0; i < 8; ++i) red[wave][lane][i] = c[i];
    __syncthreads();

    if (wave == 0) {
        float acc[8];
#pragma unroll
        for (int i = 0; i < 8; ++i) {
            float s = red[0][lane][i];
#pragma unroll
            for (int w = 1; w < 8; ++w) s += red[w][lane][i];
            acc[i] = s;
        }
        // Write logits: row rr needs C[rr][rr % 6]; exactly one lane holds it.
#pragma unroll
        for (int vr = 0; vr < 8; ++vr) {
            const int M    = vr + 8 * half;
            const int rr   = tile * 16 + M;
            const int kcol = rr % Kn;
            if (m == kcol) {
                const int bbb = rr / Kn;
                out[rr] = hasv[bbb] ? (acc[vr] + bias[kcol]) : 0.0f;
            }
        }
    }
}

// ---------------------------------------------------------------------------
extern "C" void kernel_launch(void* const* d_in, const int* in_sizes, int n_in,
                              void* d_out, int out_size, void* d_ws, size_t ws_size,
                              hipStream_t stream) {
    const float* hs    = (const float*)d_in[0];
    const float* W     = (const float*)d_in[1];
    const float* bias  = (const float*)d_in[2];
    const int*   ids   = (const int*)d_in[3];
    const int*   amask = (const int*)d_in[4];
    const int*   tok   = (const int*)d_in[5];
    float*       out   = (float*)d_out;

    int* pos  = (int*)d_ws;       // 192 ints
    int* hasv = pos + NROWS;      // 32 ints

    k_positions<<<Bn,    256, 0, stream>>>(ids, amask, tok, pos, hasv);
    k_logits   <<<TILES, 256, 0, stream>>>(hs, W, bias, pos, hasv, out);
}